// UsertoItemScorer_alone_57913339020025
// MI455X (gfx1250) — compile-verified
//
#include <hip/hip_runtime.h>

// Reference:
//   playlist_h = mean(track_emb[sampled_tracks], axis=1)            [P, 64]
//   pair_score = sum(playlist_h[edge_src] * track_emb[edge_dst])    [E]
//
// Strategy: 16 edges per wave32. Chain 16x V_WMMA_F32_16X16X4_F32
// (f32 in / f32 out, matches reference precision) accumulating
// D = A(16x64) x B(64x16); take the diagonal.  A rows are playlist sums
// (a0+a1) computed on the fly; the *0.5 mean factor is linear so it is
// hoisted out of the WMMA chain and applied once to the final diagonal
// value.  All operand loads are contiguous 8-byte float2 gathers matching
// the WMMA f32 A/B VGPR layouts:
//   A 16x4: lane m -> A[m][2*(lane>>4) + {0,1}]  (VGPR0,VGPR1)
//   B 4x16: lane n -> B[2*(lane>>4) + {0,1}][n]  (VGPR0,VGPR1)
// so per K-chunk each lane does float2 loads at offset 4*kk + 2*(lane>>4);
// every embedding byte is read exactly once per wave.

typedef __attribute__((ext_vector_type(2))) float v2f;
typedef __attribute__((ext_vector_type(8))) float v8f;

constexpr int EMB_D = 64;

__global__ __launch_bounds__(256) void uts_edge_score_wmma(
    const float* __restrict__ emb,   // [N_TRACKS, 64]
    const int*   __restrict__ esrc,  // [E] playlist ids
    const int*   __restrict__ edst,  // [E] track ids
    const int*   __restrict__ samp,  // [P, 2] sampled track ids
    float*       __restrict__ out,   // [E] scores
    int nE)
{
    const int lane        = threadIdx.x & 31;
    const int waveInBlock = threadIdx.x >> 5;
    const long waveId = (long)blockIdx.x * (blockDim.x >> 5) + waveInBlock;
    const long base   = waveId * 16;          // first edge of this wave's tile
    if (base >= nE) return;                   // wave-uniform exit (EXEC stays all-1 for WMMA)

    // Edge j (0..15) is owned by lanes j and j+16.
    const int j = lane & 15;
    long e = base + j;
    if (e >= nE) e = nE - 1;                  // clamp tail; stores are masked below

    const int src = esrc[e];
    const int dst = edst[e];
    const int s0  = samp[2 * src + 0];
    const int s1  = samp[2 * src + 1];

    const float* pA0 = emb + (long)s0 * EMB_D;   // playlist sample rows
    const float* pA1 = emb + (long)s1 * EMB_D;
    const float* pB  = emb + (long)dst * EMB_D;  // track row

    const int half2 = (lane >> 4) << 1;          // 0 for lanes 0-15, 2 for lanes 16-31

    v8f c = {};  // 16x16 f32 accumulator (diagonal is what we want)
#pragma unroll
    for (int kk = 0; kk < EMB_D / 4; ++kk) {
        const int off = kk * 4 + half2;
        v2f a0 = *(const v2f*)(pA0 + off);
        v2f a1 = *(const v2f*)(pA1 + off);
        v2f b  = *(const v2f*)(pB  + off);
        v2f a  = a0 + a1;                        // playlist sum; *0.5 hoisted to epilogue
        // D = A x B + C   (8 args: neg_a, A, neg_b, B, c_mod, C, reuse_a, reuse_b)
        c = __builtin_amdgcn_wmma_f32_16x16x4_f32(
                false, a, false, b, (short)0, c, false, false);
    }

    // Diagonal extraction.  C/D layout: VGPR r -> lanes 0-15: D[r][lane],
    // lanes 16-31: D[r+8][lane-16].  So D[i][i] lives at:
    //   i < 8 : VGPR i,   lane i
    //   i >= 8: VGPR i-8, lane i+16
    // Active lanes: 0-7 and 24-31; in both cases the VGPR index is (lane & 7).
    const int k = lane & 7;
    float s01   = (k & 1) ? c[1] : c[0];
    float s23   = (k & 1) ? c[3] : c[2];
    float s45   = (k & 1) ? c[5] : c[4];
    float s67   = (k & 1) ? c[7] : c[6];
    float s0123 = (k & 2) ? s23 : s01;
    float s4567 = (k & 2) ? s67 : s45;
    float val   = 0.5f * ((k & 4) ? s4567 : s0123);  // apply mean factor once

    const bool active = (lane < 8) || (lane >= 24);
    if (active) {
        const long o = base + ((lane < 16) ? lane : lane - 16);  // 0-7 and 8-15
        if (o < nE) out[o] = val;
    }
}

extern "C" void kernel_launch(void* const* d_in, const int* in_sizes, int n_in,
                              void* d_out, int out_size, void* d_ws, size_t ws_size,
                              hipStream_t stream) {
    (void)n_in; (void)d_ws; (void)ws_size; (void)out_size;
    const float* emb  = (const float*)d_in[0];   // track_emb   [N_TRACKS*64] f32
    const int*   esrc = (const int*)  d_in[1];   // edge_src    [E] i32
    const int*   edst = (const int*)  d_in[2];   // edge_dst    [E] i32
    const int*   samp = (const int*)  d_in[3];   // sampled     [P*2] i32
    float*       out  = (float*)d_out;           // pair_score  [E] f32

    const int nE = in_sizes[1];
    const long waves = ((long)nE + 15) / 16;     // 16 edges per wave32
    const int  threads = 256;                    // 8 waves per block
    const int  wavesPerBlock = threads / 32;
    const long blocks = (waves + wavesPerBlock - 1) / wavesPerBlock;

    uts_edge_score_wmma<<<(int)blocks, threads, 0, stream>>>(
        emb, esrc, edst, samp, out, nE);
}